// MLP_17308718203313
// MI455X (gfx1250) — compile-verified
//
#include <hip/hip_runtime.h>

// ---- problem constants (match reference) ----
#define NN   200000
#define DD   512
#define LL   3
#define GG   32
#define EPSV 1e-5f

// ---- tiling ----
#define ROWS  32           // rows per block (200000 % 32 == 0 -> 6250 blocks)
#define BLK   256          // 8 wave32 waves
#define KC    32           // k-chunk staged in LDS per TDM transfer (64 KB)

typedef float v2f __attribute__((ext_vector_type(2)));
typedef float v8f __attribute__((ext_vector_type(8)));
typedef int   v4i __attribute__((ext_vector_type(4)));
typedef int   v8i __attribute__((ext_vector_type(8)));

__global__ void zero_ws(float* __restrict__ p, int n) {
  int i = blockIdx.x * blockDim.x + threadIdx.x;
  if (i < n) p[i] = 0.0f;
}

// ---- Tensor Data Mover: 2D tile (KC rows x 512 cols, fp32) global -> LDS ----
// D# per CDNA5 ISA ch.8: group0 = {count/lds_addr/global_addr/type},
// group1 = {data_size, tensor dims/strides, tile dims}. 2D -> groups 2/3 NULL.
__device__ __forceinline__ void tdm_load_panel(unsigned lds_byte, const float* gptr) {
  unsigned long long ga = (unsigned long long)(uintptr_t)gptr;
  v4i g0;
  g0.x = 1;                                            // count=1, user mode
  g0.y = (int)lds_byte;                                // lds_addr (bytes)
  g0.z = (int)(unsigned)ga;                            // global_addr[31:0]
  g0.w = (int)(((unsigned)(ga >> 32) & 0x01FFFFFFu)    // global_addr[56:32]
               | 0x80000000u);                         // type=2 ("image")
  v8i g1;
  g1[0] = 0x00020000;            // workgroup_mask=0 | data_size=2 (4 bytes)
  g1[1] = (int)(512u << 16);     // tensor_dim0[15:0]=512 (cols)
  g1[2] = (int)(512u << 16);     // tensor_dim0 hi=0 | tensor_dim1[15:0]=512 (rows avail)
  g1[3] = (int)(512u << 16);     // tensor_dim1 hi=0 | tile_dim0=512
  g1[4] = KC;                    // tile_dim1=KC rows, tile_dim2=0
  g1[5] = 512;                   // tensor_dim0_stride=512 elements
  g1[6] = 0;
  g1[7] = 0;
  asm volatile("tensor_load_to_lds %0, %1" :: "s"(g0), "s"(g1) : "memory");
}

__launch_bounds__(BLK)
__global__ void fused_gnn(const float* __restrict__ X,      // [N, D]
                          const int*   __restrict__ batch,  // [N] sorted
                          const float* __restrict__ Wc,     // [L, D, D]
                          const float* __restrict__ bc,     // [L, D]
                          const float* __restrict__ gamma,  // [L, D]
                          const float* __restrict__ beta,   // [L, D]
                          const float* __restrict__ avec,   // [L]
                          float* __restrict__ cnt,          // [G]
                          float* __restrict__ pool)         // [L+1, G, D] (sums)
{
  extern __shared__ float smem[];
  float* xcur = smem;                  // ROWS x D   (residual stream)
  float* hbuf = smem + ROWS * DD;      // ROWS x D   (pre-LN activations)
  float* wbuf = smem + 2 * ROWS * DD;  // 2 x KC x D (double-buffered W panel)

  const int tid     = threadIdx.x;
  const int lane    = tid & 31;
  const int wave    = tid >> 5;
  const int rowBase = blockIdx.x * ROWS;

  // GEMM role of this wave: 2 row-groups x 4 col-quarters
  const int rg   = wave >> 2;        // rows [16*rg, 16*rg+16)
  const int cq   = wave & 3;         // cols [128*cq, 128*cq+128)
  const int R    = rg * 16;
  const int C0   = cq * 128;
  const int half = lane >> 4;        // lanes 16-31 carry K+2/K+3 fragments
  const int m    = lane & 15;

  const unsigned wbuf_byte0 = (unsigned)(uintptr_t)wbuf; // LDS offset = addr[31:0]

  // ---- load x tile into LDS, coalesced 16B ----
  {
    const float4* Xv = (const float4*)(X + (size_t)rowBase * DD);
    float4* xv = (float4*)xcur;
    #pragma unroll 4
    for (int i = tid; i < ROWS * DD / 4; i += BLK) xv[i] = Xv[i];
  }

  // ---- per-graph node counts: short run-length loop (batch is sorted) ----
  if (tid == 0) {
    int r = 0;
    while (r < ROWS) {
      int g = batch[rowBase + r];
      int run = 1;
      while (r + run < ROWS && batch[rowBase + r + run] == g) run++;
      atomicAdd(&cnt[g], (float)run);
      r += run;
    }
  }
  __syncthreads();

  // ---- pool level 0: segment-sum of raw x (wave -> 4 rows, lane -> 16 cols) ----
  {
    const int c0 = lane * 16;
    float pacc[16];
    #pragma unroll
    for (int i = 0; i < 16; ++i) pacc[i] = 0.0f;
    int curg = batch[rowBase + wave * 4];
    for (int rr = 0; rr < 4; ++rr) {
      const int row = wave * 4 + rr;
      const int g = batch[rowBase + row];
      if (g != curg) {
        #pragma unroll
        for (int i = 0; i < 16; ++i) {
          atomicAdd(&pool[(size_t)curg * DD + c0 + i], pacc[i]);
          pacc[i] = 0.0f;
        }
        curg = g;
      }
      #pragma unroll
      for (int i = 0; i < 16; ++i) pacc[i] += xcur[row * DD + c0 + i];
    }
    #pragma unroll
    for (int i = 0; i < 16; ++i)
      atomicAdd(&pool[(size_t)curg * DD + c0 + i], pacc[i]);
  }

  // ---- layer loop: fully fused, x tile stays in LDS ----
  const int NCHUNK = DD / KC;   // 16
  for (int l = 0; l < LL; ++l) {
    const float* Wl = Wc + (size_t)l * DD * DD;
    const float* bl = bc    + l * DD;
    const float* gl = gamma + l * DD;
    const float* be = beta  + l * DD;
    const float  al = avec[l];
    float* poolL = pool + (size_t)(l + 1) * GG * DD;

    v8f acc[8] = {};
    const float2* ap = (const float2*)(xcur + (R + m) * DD);

    // prime TDM pipeline: panel 0 -> wbuf[0]
    if (wave == 0) {
      tdm_load_panel(wbuf_byte0, Wl);
      __builtin_amdgcn_s_wait_tensorcnt(0);
    }
    __syncthreads();

    #pragma unroll 1
    for (int c = 0; c < NCHUNK; ++c) {
      // async prefetch of next panel into the other buffer (overlaps compute)
      if (wave == 0 && c + 1 < NCHUNK)
        tdm_load_panel(wbuf_byte0 + (unsigned)(((c + 1) & 1) * (KC * DD * 4)),
                       Wl + (size_t)(c + 1) * KC * DD);

      // B fragments from the staged panel (immediate-offset ds loads)
      const float* wbl = wbuf + (c & 1) * (KC * DD) + (2 * half) * DD + C0 + m;
      const int kbase = c * KC;

      #pragma unroll
      for (int kk = 0; kk < KC; kk += 4) {
        const int k0 = kbase + kk;
        float2 av = ap[(k0 >> 1) + half];   // A 16x4 f32 fragment (ds_load_b64)
        v2f a = {av.x, av.y};
        #pragma unroll
        for (int j = 0; j < 8; ++j) {
          v2f b = {wbl[kk * DD + j * 16], wbl[kk * DD + j * 16 + DD]};
          acc[j] = __builtin_amdgcn_wmma_f32_16x16x4_f32(
              /*neg_a=*/false, a, /*neg_b=*/false, b,
              /*c_mod=*/(short)0, acc[j], /*reuse_a=*/false, /*reuse_b=*/false);
        }
      }

      if (wave == 0 && c + 1 < NCHUNK) __builtin_amdgcn_s_wait_tensorcnt(0);
      __syncthreads();   // panel (c+1) visible; reads of panel c complete
    }

    // store h = acc + bias to LDS (C/D layout: VGPR v -> rows v / v+8)
    #pragma unroll
    for (int j = 0; j < 8; ++j) {
      const int n = C0 + j * 16 + m;
      const float bias = bl[n];
      #pragma unroll
      for (int v = 0; v < 8; ++v)
        hbuf[(R + v + 8 * half) * DD + n] = acc[j][v] + bias;
    }
    __syncthreads();

    // ---- LayerNorm + PReLU + pooling + residual (wave -> 4 rows) ----
    {
      const int c0 = lane * 16;
      float pacc[16];
      #pragma unroll
      for (int i = 0; i < 16; ++i) pacc[i] = 0.0f;
      int curg = batch[rowBase + wave * 4];

      for (int rr = 0; rr < 4; ++rr) {
        const int row = wave * 4 + rr;
        float hv[16];
        float s = 0.0f, s2 = 0.0f;
        #pragma unroll
        for (int i = 0; i < 16; ++i) {
          float v = hbuf[row * DD + c0 + i];
          hv[i] = v; s += v; s2 += v * v;
        }
        #pragma unroll
        for (int off = 16; off > 0; off >>= 1) {
          s  += __shfl_xor(s,  off, 32);
          s2 += __shfl_xor(s2, off, 32);
        }
        const float mean = s * (1.0f / DD);
        const float var  = s2 * (1.0f / DD) - mean * mean;
        const float rstd = rsqrtf(var + EPSV);

        const int g = batch[rowBase + row];
        if (g != curg) {
          #pragma unroll
          for (int i = 0; i < 16; ++i) {
            atomicAdd(&poolL[(size_t)curg * DD + c0 + i], pacc[i]);
            pacc[i] = 0.0f;
          }
          curg = g;
        }
        #pragma unroll
        for (int i = 0; i < 16; ++i) {
          const int c = c0 + i;
          float hn = (hv[i] - mean) * rstd * gl[c] + be[c];
          float hp = hn >= 0.0f ? hn : al * hn;
          pacc[i] += hp;
          xcur[row * DD + c] += hp;   // residual -> input of next layer
        }
      }
      #pragma unroll
      for (int i = 0; i < 16; ++i)
        atomicAdd(&poolL[(size_t)curg * DD + c0 + i], pacc[i]);
    }
    __syncthreads();
  }
}

__launch_bounds__(256)
__global__ void risk_kernel(const float* __restrict__ pool,  // [L+1, G, D] sums
                            const float* __restrict__ cnt,   // [G]
                            const float* __restrict__ Wr,    // [D*(L+1)]
                            const float* __restrict__ br,    // [1]
                            float* __restrict__ out)         // [G]
{
  const int g = blockIdx.x;
  const float invc = 1.0f / fmaxf(cnt[g], 1.0f);
  float partial = 0.0f;
  for (int f = threadIdx.x; f < DD * (LL + 1); f += 256) {
    const int lvl = f >> 9;
    const int n   = f & (DD - 1);
    partial += pool[(size_t)lvl * GG * DD + (size_t)g * DD + n] * invc * Wr[f];
  }
  #pragma unroll
  for (int off = 16; off > 0; off >>= 1) partial += __shfl_xor(partial, off, 32);
  __shared__ float red[8];
  if ((threadIdx.x & 31) == 0) red[threadIdx.x >> 5] = partial;
  __syncthreads();
  if (threadIdx.x == 0) {
    float t = 0.0f;
    #pragma unroll
    for (int i = 0; i < 8; ++i) t += red[i];
    out[g] = t + br[0];
  }
}

extern "C" void kernel_launch(void* const* d_in, const int* in_sizes, int n_in,
                              void* d_out, int out_size, void* d_ws, size_t ws_size,
                              hipStream_t stream) {
  (void)in_sizes; (void)n_in; (void)out_size; (void)ws_size;
  const float* X     = (const float*)d_in[0];
  const int*   batch = (const int*)  d_in[1];
  const float* Wc    = (const float*)d_in[2];
  const float* bc    = (const float*)d_in[3];
  const float* gam   = (const float*)d_in[4];
  const float* bet   = (const float*)d_in[5];
  const float* avec  = (const float*)d_in[6];
  const float* Wr    = (const float*)d_in[7];
  const float* br    = (const float*)d_in[8];
  float* out = (float*)d_out;

  // workspace: [cnt(G)] [pool (L+1)*G*D]
  float* cnt  = (float*)d_ws;
  float* pool = cnt + GG;
  const int zn = GG + (LL + 1) * GG * DD;

  zero_ws<<<(zn + 255) / 256, 256, 0, stream>>>(cnt, zn);

  // x-tile (64K) + h-buf (64K) + double-buffered W panel (128K) = 256 KB / WGP
  const size_t lds_bytes = (size_t)(2 * ROWS * DD + 2 * KC * DD) * sizeof(float);
  hipFuncSetAttribute((const void*)fused_gnn,
                      hipFuncAttributeMaxDynamicSharedMemorySize, (int)lds_bytes);
  fused_gnn<<<NN / ROWS, BLK, lds_bytes, stream>>>(X, batch, Wc, bc, gam, bet,
                                                   avec, cnt, pool);

  risk_kernel<<<GG, 256, 0, stream>>>(pool, cnt, Wr, br, out);
}